// CRF_87522843559635
// MI455X (gfx1250) — compile-verified
//
#include <hip/hip_runtime.h>
#include <cstdint>

// CRF Viterbi decode + one-hot, fused single kernel for gfx1250 (MI455X).
//
// Layout: one wave32 per batch row, lane = tag index j (24 of 32 lanes live).
//  - transitions column T[.][j] cached in 24 VGPRs per lane (via LDS stage)
//  - alpha broadcast via v_readlane_b32 (immediate lanes, unrolled)
//  - argmax reduction split into 4 independent chains (ILP / VOPD friendly),
//    merged low-range-first with strict '>' => exact JAX first-max tie-break
//  - backpointers kept in LDS (511*24 bytes per batch) -> zero global scratch
//  - emissions streamed through an 8-deep LDS ring via CDNA5 async DMA
//    (global_load_async_to_lds_b32, gated by s_wait_asynccnt 6: async loads
//    complete in order, so <=6 outstanding => step t's buffer has landed)
//  - backtrack walked redundantly by all lanes (uniform LDS broadcast reads)
//    so each step writes one coalesced 96B one-hot row.

#define NUM_TAGS 24
#define SEQ_LEN  512
#define BATCH    8192

#define WAVES_PER_BLOCK 4
#define THREADS_PER_BLOCK (WAVES_PER_BLOCK * 32)
#define BP_PAD 12288                       // padded (SEQ_LEN-1)*NUM_TAGS
#define EMIT_BUFS 8                        // async pipeline depth

__device__ __forceinline__ float read_lane_f32(float v, int lane) {
  return __int_as_float(__builtin_amdgcn_readlane(__float_as_int(v), lane));
}

__device__ __forceinline__ void async_load_b32_to_lds(uint32_t lds_byte_addr,
                                                      const float* gaddr) {
  // CDNA5 async DMA: copies the 4-byte word at global address gaddr to LDS byte address lds_byte_addr (per active lane),
  // tracked by ASYNCcnt. GV addressing (64-bit VGPR address, saddr=off).
  uint64_t a = (uint64_t)(uintptr_t)gaddr;
  asm volatile("global_load_async_to_lds_b32 %0, %1, off"
               :
               : "v"(lds_byte_addr), "v"(a)
               : "memory");
}

__device__ __forceinline__ void upd_argmax(float& best, int& bidx,
                                           float cand, int i) {
  bool gt = cand > best;          // strict '>' keeps earliest index on ties
  best = gt ? cand : best;
  bidx = gt ? i : bidx;
}

__global__ __launch_bounds__(THREADS_PER_BLOCK)
void crf_viterbi_onehot(const float* __restrict__ inp,     // [B, T, K] f32
                        const float* __restrict__ trans,   // [K, K] f32
                        float* __restrict__ out)           // [B, T, K] f32
{
  __shared__ float s_trans[NUM_TAGS * NUM_TAGS];                 // 2304 B
  __shared__ float s_emit[WAVES_PER_BLOCK * EMIT_BUFS * 32];     // 4096 B
  __shared__ unsigned char s_bp[WAVES_PER_BLOCK * BP_PAD];       // 49152 B

  const int tid  = threadIdx.x;
  const int w    = tid >> 5;     // wave in block
  const int lane = tid & 31;     // tag index j (lanes 24..31 inactive)
  const long b   = (long)blockIdx.x * WAVES_PER_BLOCK + w;  // batch row

  // Stage transitions to LDS (coalesced), then pull my column into VGPRs.
  for (int i = tid; i < NUM_TAGS * NUM_TAGS; i += THREADS_PER_BLOCK)
    s_trans[i] = trans[i];
  __syncthreads();

  const int jj = (lane < NUM_TAGS) ? lane : (NUM_TAGS - 1);  // clamped lane
  float tc[NUM_TAGS];  // tc[i] = trans[i][j]
#pragma unroll
  for (int i = 0; i < NUM_TAGS; ++i) tc[i] = s_trans[i * NUM_TAGS + jj];

  const float* emit_base = inp + (size_t)b * SEQ_LEN * NUM_TAGS;
  float alpha = emit_base[jj];   // alpha init = emissions[b][0][j]

  float* semit_wave = &s_emit[w * (EMIT_BUFS * 32)];
  unsigned char* bp = &s_bp[w * BP_PAD];

  // Pre-issue async DMA for steps 1..7 (ring buffers 1..7).
#pragma unroll
  for (int p = 1; p <= EMIT_BUFS - 1; ++p) {
    async_load_b32_to_lds(
        (uint32_t)(uintptr_t)&semit_wave[(p & (EMIT_BUFS - 1)) * 32 + lane],
        emit_base + (size_t)p * NUM_TAGS + jj);
  }

  // ---- forward Viterbi scan ----
  for (int t = 1; t < SEQ_LEN; ++t) {
    // Async loads complete in order: <= EMIT_BUFS-2 outstanding means the
    // load for step t (oldest in flight) has landed in LDS.
    asm volatile("s_wait_asynccnt 6" ::: "memory");
    float emit = semit_wave[(t & (EMIT_BUFS - 1)) * 32 + lane];

    if (t + EMIT_BUFS - 1 < SEQ_LEN) {   // keep the ring full
      async_load_b32_to_lds(
          (uint32_t)(uintptr_t)
              &semit_wave[((t + EMIT_BUFS - 1) & (EMIT_BUFS - 1)) * 32 + lane],
          emit_base + (size_t)(t + EMIT_BUFS - 1) * NUM_TAGS + jj);
    }

    // best_j = max_i(alpha[i] + T[i][j]) with argmax; 4 independent chains
    // over contiguous ranges, merged low-range-first (exact first-max ties).
    float b0 = read_lane_f32(alpha, 0)  + tc[0];  int i0 = 0;
    float b1 = read_lane_f32(alpha, 6)  + tc[6];  int i1 = 6;
    float b2 = read_lane_f32(alpha, 12) + tc[12]; int i2 = 12;
    float b3 = read_lane_f32(alpha, 18) + tc[18]; int i3 = 18;
#pragma unroll
    for (int i = 1; i < 6; ++i) {
      upd_argmax(b0, i0, read_lane_f32(alpha, i)      + tc[i],      i);
      upd_argmax(b1, i1, read_lane_f32(alpha, 6 + i)  + tc[6 + i],  6 + i);
      upd_argmax(b2, i2, read_lane_f32(alpha, 12 + i) + tc[12 + i], 12 + i);
      upd_argmax(b3, i3, read_lane_f32(alpha, 18 + i) + tc[18 + i], 18 + i);
    }
    upd_argmax(b0, i0, b1, i1);   // (0..5) vs (6..11): tie -> lower range
    upd_argmax(b2, i2, b3, i3);   // (12..17) vs (18..23)
    upd_argmax(b0, i0, b2, i2);   // merge halves
    // NB: upd_argmax(best, idx, candBest, candIdx) works for merging since
    // it is exactly "take candidate only if strictly greater".

    if (lane < NUM_TAGS)
      bp[(t - 1) * NUM_TAGS + lane] = (unsigned char)i0;

    alpha = b0 + emit;
  }

  // ---- last tag: argmax over final alpha (uniform across lanes) ----
  float bv = read_lane_f32(alpha, 0);
  int tag = 0;
#pragma unroll
  for (int i = 1; i < NUM_TAGS; ++i) {
    float ai = read_lane_f32(alpha, i);
    bool gt = ai > bv;
    bv = gt ? ai : bv;
    tag = gt ? i : tag;
  }

  // ---- backtrack + one-hot output (coalesced 96B row per step) ----
  float* ob = out + (size_t)b * SEQ_LEN * NUM_TAGS;
  if (lane < NUM_TAGS)
    ob[(size_t)(SEQ_LEN - 1) * NUM_TAGS + lane] = (lane == tag) ? 1.0f : 0.0f;

  for (int t = SEQ_LEN - 2; t >= 0; --t) {
    tag = bp[t * NUM_TAGS + tag];  // uniform address -> LDS broadcast read
    if (lane < NUM_TAGS)
      ob[(size_t)t * NUM_TAGS + lane] = (lane == tag) ? 1.0f : 0.0f;
  }
}

extern "C" void kernel_launch(void* const* d_in, const int* in_sizes, int n_in,
                              void* d_out, int out_size, void* d_ws, size_t ws_size,
                              hipStream_t stream) {
  (void)in_sizes; (void)n_in; (void)out_size; (void)d_ws; (void)ws_size;
  const float* inp   = (const float*)d_in[0];  // [8192, 512, 24] f32
  const float* trans = (const float*)d_in[1];  // [24, 24] f32
  float* out = (float*)d_out;                  // [8192, 512, 24] f32

  dim3 block(THREADS_PER_BLOCK);
  dim3 grid(BATCH / WAVES_PER_BLOCK);  // one wave per batch row
  crf_viterbi_onehot<<<grid, block, 0, stream>>>(inp, trans, out);
}